// HSMMModel_54657753808971
// MI455X (gfx1250) — compile-verified
//
#include <hip/hip_runtime.h>
#include <hip/hip_bf16.h>

#define HSMM_B 32
#define HSMM_T 256
#define HSMM_Z 128
#define HSMM_VOCAB 30000
#define NEG_INF (-1.0e6f)

typedef __attribute__((ext_vector_type(2))) float v2f;
typedef __attribute__((ext_vector_type(8))) float v8f;

// ---------------------------------------------------------------------------
// Phase 1a: per-block partial online-LSE over the vocab axis of each emission
// matrix. Grid: (32 row-blocks, 4 matrices). Block: 256 threads = 2 rows x 128
// columns, fully coalesced (row-major [30000,128] reads).
// ---------------------------------------------------------------------------
__global__ void col_lse_partial(const float* __restrict__ e2,
                                const float* __restrict__ e3,
                                const float* __restrict__ e4,
                                const float* __restrict__ e5,
                                float* __restrict__ part) {
  const int n = blockIdx.y;
  const int blk = blockIdx.x;
  const float* em = (n == 0) ? e2 : (n == 1) ? e3 : (n == 2) ? e4 : e5;
  const int tid = threadIdx.x;
  const int z = tid & 127;
  const int rr = tid >> 7;  // 0 or 1

  const int rows = 938;  // ceil(30000/32)
  int v0 = blk * rows + rr;
  int vend = (blk + 1) * rows;
  if (vend > HSMM_VOCAB) vend = HSMM_VOCAB;

  float m = -3.0e38f, s = 0.0f;
  for (int v = v0; v < vend; v += 2) {
    float x = em[v * HSMM_Z + z];
    float nm = fmaxf(m, x);
    s = s * __expf(m - nm) + __expf(x - nm);
    m = nm;
  }

  __shared__ float sm_[256];
  __shared__ float ss_[256];
  sm_[tid] = m;
  ss_[tid] = s;
  __syncthreads();
  if (rr == 0) {
    float m1 = sm_[tid + 128], s1 = ss_[tid + 128];
    float M = fmaxf(m, m1);
    float S = s * __expf(m - M) + s1 * __expf(m1 - M);
    size_t o = ((size_t)(n * 32 + blk) * HSMM_Z + z) * 2;
    part[o + 0] = M;
    part[o + 1] = S;
  }
}

// Phase 1b: combine the 32 partials per (matrix, column) -> colLSE[n][z].
__global__ void col_lse_final(const float* __restrict__ part,
                              float* __restrict__ colLSE) {
  const int n = blockIdx.x;
  const int z = threadIdx.x;
  float M = -3.0e38f, S = 0.0f;
  for (int blk = 0; blk < 32; ++blk) {
    size_t o = ((size_t)(n * 32 + blk) * HSMM_Z + z) * 2;
    float m = part[o + 0], s = part[o + 1];
    float nM = fmaxf(M, m);
    S = S * __expf(M - nM) + s * __expf(m - nM);
    M = nM;
  }
  colLSE[n * HSMM_Z + z] = M + logf(S);
}

// ---------------------------------------------------------------------------
// Phase 1c: expT[z'][z] = softmax(transition, axis=-1) = exp(log_softmax).
// One block per row.
// ---------------------------------------------------------------------------
__global__ void row_softmax(const float* __restrict__ trans,
                            float* __restrict__ expT) {
  const int r = blockIdx.x;
  const int c = threadIdx.x;
  float x = trans[r * HSMM_Z + c];
  __shared__ float red[HSMM_Z];
  red[c] = x;
  __syncthreads();
  for (int off = 64; off > 0; off >>= 1) {
    if (c < off) red[c] = fmaxf(red[c], red[c + off]);
    __syncthreads();
  }
  float m = red[0];
  __syncthreads();
  float e = __expf(x - m);
  red[c] = e;
  __syncthreads();
  for (int off = 64; off > 0; off >>= 1) {
    if (c < off) red[c] += red[c + off];
    __syncthreads();
  }
  float S = red[0];
  expT[r * HSMM_Z + c] = e / S;
}

// Phase 1d: length-emission log-softmax over K (NEG_INF slots for n<2 vanish
// exactly in f32, matching the reference). lemls[z*4 + (n-2)] for n=2..5.
__global__ void lem_logsoftmax(const float* __restrict__ lem,
                               float* __restrict__ lemls) {
  const int z = threadIdx.x;
  float v0 = lem[z * 4 + 0], v1 = lem[z * 4 + 1];
  float v2 = lem[z * 4 + 2], v3 = lem[z * 4 + 3];
  float m = fmaxf(fmaxf(v0, v1), fmaxf(v2, v3));
  float s = __expf(v0 - m) + __expf(v1 - m) + __expf(v2 - m) + __expf(v3 - m);
  float lse = m + logf(s);
  lemls[z * 4 + 0] = v0 - lse;
  lemls[z * 4 + 1] = v1 - lse;
  lemls[z * 4 + 2] = v2 - lse;
  lemls[z * 4 + 3] = v3 - lse;
}

// ---------------------------------------------------------------------------
// DP kernel helpers
// ---------------------------------------------------------------------------
__device__ __forceinline__ int ring6(int r) {  // r in [0, 11]
  return (r >= 6) ? (r - 6) : r;
}

__device__ __forceinline__ float load_e(const int* __restrict__ sub,
                                        const float* __restrict__ em,
                                        const float* __restrict__ colLSE_n,
                                        const float* __restrict__ lemls,
                                        int n, int b, int t, int z) {
  int s = t + 1 - n;
  if (s < 0) return NEG_INF;
  int L = HSMM_T - n + 1;
  int id = sub[b * L + s];
  if (id >= HSMM_VOCAB) id = 1;  // -> UNK
  float lm = lemls[z * 4 + (n - 2)];
  if (id == 0) return lm;             // PAD: emission score forced to 0
  if (id == 1) return NEG_INF + lm;   // UNK: emission score forced to NEG_INF
  return em[id * HSMM_Z + z] - colLSE_n[z] + lm;
}

// ---------------------------------------------------------------------------
// Semi-Markov forward scan. One workgroup per batch element, 8 waves.
// Per step: ta[j][z] = m_j + log( exp(W[j]-m_j) . expT )  via f32 WMMA,
// then alpha[z] = LSE_{n=2..5}( ta[n-1][z] + E_t[n][z] ).
//
// WMMA lane maps (CDNA5 ISA 7.12.2, wave32):
//   A f32 16x4  : lane L -> M=L%16 ; VGPR p holds K = p + 2*(L>=16)
//   B f32 4x16  : lane L -> N=L%16 ; VGPR p holds K = p + 2*(L>=16)
//   C/D f32 16x16: lane L -> N=L%16 ; VGPR r holds M = r + 8*(L>=16)
//
// A and B live in LDS in "K-pair interleaved" layouts so each fragment
// register pair is a single 8-byte-aligned ds_load_b64:
//   sA2[(k>>1)*32  + m*2 + (k&1)] = A^T
//   sT2[(k>>1)*256 + n*2 + (k&1)] = expT
// ---------------------------------------------------------------------------
__global__ __launch_bounds__(256) void hsmm_dp(
    const int* __restrict__ xlen,
    const int* __restrict__ sub2, const int* __restrict__ sub3,
    const int* __restrict__ sub4, const int* __restrict__ sub5,
    const float* __restrict__ em2, const float* __restrict__ em3,
    const float* __restrict__ em4, const float* __restrict__ em5,
    const float* __restrict__ expT_g, const float* __restrict__ colLSE,
    const float* __restrict__ lemls, float* __restrict__ out) {
  extern __shared__ float smem[];
  float* sT2 = smem;               // [64][128][2] expT pair-interleaved, 64 KB
  float* sA2 = sT2 + 128 * 128;    // [64][16][2]  A^T pair-interleaved
  float* sP = sA2 + 128 * 16;      // [4][128]     matmul result rows j=1..4
  float* cw = sP + 4 * 128;        // [6][128]     circular alpha history
  float* sm_ = cw + 6 * 128;       // [8]          per-row maxima
  float* red = sm_ + 8;            // [128]        final-LSE scratch

  const int b = blockIdx.x;
  const int tid = threadIdx.x;
  const int lane = tid & 31;
  const int wv = tid >> 5;

  for (int i = tid; i < 128 * 128; i += 256) {
    int k = i >> 7, n = i & 127;
    sT2[((k >> 1) << 8) + n * 2 + (k & 1)] = expT_g[i];
  }
  for (int i = tid; i < 6 * 128; i += 256) cw[i] = (i == 0) ? 0.0f : NEG_INF;
  __syncthreads();

  const int len = xlen[b];
  int h = 0;  // cw[ring6(h+j)] == logical W[j] == alpha[t-1-j]

  for (int t = 0; t < HSMM_T; ++t) {
    const int newh = (h == 0) ? 5 : (h - 1);

    float e2v = NEG_INF, e3v = NEG_INF, e4v = NEG_INF, e5v = NEG_INF;
    if (tid < 128) {
      // Prefetch emission terms early; consumed only in phase 4.
      e2v = load_e(sub2, em2, colLSE + 0 * HSMM_Z, lemls, 2, b, t, tid);
      e3v = load_e(sub3, em3, colLSE + 1 * HSMM_Z, lemls, 3, b, t, tid);
      e4v = load_e(sub4, em4, colLSE + 2 * HSMM_Z, lemls, 4, b, t, tid);
      e5v = load_e(sub5, em5, colLSE + 3 * HSMM_Z, lemls, 5, b, t, tid);

      // Phase 1: m_j = max_z W[j][z] for j = 1..4 (wave j-1 per row).
      int j = 1 + (tid >> 5);
      const float* wr = &cw[ring6(h + j) * HSMM_Z];
      float m = fmaxf(fmaxf(wr[lane], wr[lane + 32]),
                      fmaxf(wr[lane + 64], wr[lane + 96]));
#pragma unroll
      for (int off = 16; off > 0; off >>= 1) m = fmaxf(m, __shfl_xor(m, off, 32));
      if (lane == 0) sm_[j] = m;
    }
    __syncthreads();

    // Phase 2: sA2 = exp(W[r+1][k] - m_{r+1}) for r<4, else 0 (pair layout).
    for (int i = tid; i < 128 * 16; i += 256) {
      int k = i >> 4, r = i & 15;
      float v = 0.0f;
      if (r < 4) v = __expf(cw[ring6(h + r + 1) * HSMM_Z + k] - sm_[r + 1]);
      sA2[((k >> 1) << 5) + r * 2 + (k & 1)] = v;
    }
    __syncthreads();

    // Phase 3: P = A x expT, each wave owns 16 output columns. All lanes
    // active (EXEC all-ones as WMMA requires). One ds_load_b64 per fragment.
    {
      const int mrow = lane & 15;
      const int hi = lane >> 4;
      const int ncol = wv * 16 + mrow;
      v8f c = {};
#pragma unroll
      for (int kk = 0; kk < 128; kk += 4) {
        const int p2 = (kk >> 1) + hi;  // K-pair index (K = 2*p2 + q)
        v2f a = *(const v2f*)&sA2[(p2 << 5) + mrow * 2];
        v2f bb = *(const v2f*)&sT2[(p2 << 8) + ncol * 2];
        c = __builtin_amdgcn_wmma_f32_16x16x4_f32(
            false, a, false, bb, (short)0, c, false, false);
      }
      if (hi == 0) {  // rows 0..3 (j = 1..4) are all we need
#pragma unroll
        for (int r = 0; r < 4; ++r) sP[r * HSMM_Z + ncol] = c[r];
      }
    }
    __syncthreads();

    // Phase 4: alpha_t[z] = LSE_{n=2..5}( m_{n-1} + log P[n-1][z] + E_t[n][z] )
    if (tid < 128) {
      float ev[4] = {e2v, e3v, e4v, e5v};
      float v[4];
#pragma unroll
      for (int j = 1; j <= 4; ++j) {
        // P >= exp(0) * softmax_entry > 0 always; clamp keeps it branchless.
        float p = fmaxf(sP[(j - 1) * HSMM_Z + tid], 1.0e-37f);
        v[j - 1] = sm_[j] + __logf(p) + ev[j - 1];
      }
      float m4 = fmaxf(fmaxf(v[0], v[1]), fmaxf(v[2], v[3]));
      float s4 = __expf(v[0] - m4) + __expf(v[1] - m4) +
                 __expf(v[2] - m4) + __expf(v[3] - m4);
      float alpha = m4 + __logf(s4);
      cw[newh * HSMM_Z + tid] = alpha;
      red[tid] = alpha;
    }
    __syncthreads();

    if (t + 1 == len) {  // uniform branch per block
      if (tid < 32) {
        float m = fmaxf(fmaxf(red[tid], red[tid + 32]),
                        fmaxf(red[tid + 64], red[tid + 96]));
#pragma unroll
        for (int off = 16; off > 0; off >>= 1) m = fmaxf(m, __shfl_xor(m, off, 32));
        float s = __expf(red[tid] - m) + __expf(red[tid + 32] - m) +
                  __expf(red[tid + 64] - m) + __expf(red[tid + 96] - m);
#pragma unroll
        for (int off = 16; off > 0; off >>= 1) s += __shfl_xor(s, off, 32);
        if (tid == 0) out[b] = m + __logf(s);
      }
      __syncthreads();
    }

    h = newh;
  }
}

// ---------------------------------------------------------------------------
// Launch
// ---------------------------------------------------------------------------
extern "C" void kernel_launch(void* const* d_in, const int* in_sizes, int n_in,
                              void* d_out, int out_size, void* d_ws,
                              size_t ws_size, hipStream_t stream) {
  const int* x_lengths = (const int*)d_in[1];
  const int* sub2 = (const int*)d_in[2];
  const int* sub3 = (const int*)d_in[3];
  const int* sub4 = (const int*)d_in[4];
  const int* sub5 = (const int*)d_in[5];
  const float* trans = (const float*)d_in[6];
  const float* lem = (const float*)d_in[7];
  const float* em2 = (const float*)d_in[8];
  const float* em3 = (const float*)d_in[9];
  const float* em4 = (const float*)d_in[10];
  const float* em5 = (const float*)d_in[11];
  float* out = (float*)d_out;

  float* ws = (float*)d_ws;
  float* expT = ws;                  // 16384 floats
  float* colLSE = expT + 16384;      // 512
  float* lemls = colLSE + 512;       // 512
  float* part = lemls + 512;         // 4*32*128*2 = 32768

  col_lse_partial<<<dim3(32, 4), 256, 0, stream>>>(em2, em3, em4, em5, part);
  col_lse_final<<<4, 128, 0, stream>>>(part, colLSE);
  row_softmax<<<HSMM_Z, HSMM_Z, 0, stream>>>(trans, expT);
  lem_logsoftmax<<<1, HSMM_Z, 0, stream>>>(lem, lemls);

  const size_t shmem =
      (128 * 128 + 128 * 16 + 4 * 128 + 6 * 128 + 8 + 128) * sizeof(float);
  hsmm_dp<<<HSMM_B, 256, shmem, stream>>>(x_lengths, sub2, sub3, sub4, sub5,
                                          em2, em3, em4, em5, expT, colLSE,
                                          lemls, out);
}